// SCAESuite_2499670966426
// MI455X (gfx1250) — compile-verified
//
#include <hip/hip_runtime.h>
#include <hip/hip_bf16.h>

// Problem constants (fixed by the reference)
#define NB 1024   // tokens
#define ND 768    // activation dim (K of the GEMMs)
#define NF 24576  // dictionary size
#define NK 64     // top-k
#define NC 32     // connections per feature

typedef __attribute__((ext_vector_type(2))) float v2f;
typedef __attribute__((ext_vector_type(8))) float v8f;
// The async-LDS builtin wants pointers to a *generic* GCC vector type
// 'int __attribute__((vector_size(16)))' (per the round-2 diagnostic).
typedef int v4i_raw __attribute__((vector_size(16)));

// LDS row stride for GEMM tiles: 64 + 4 pad.
// 68 floats = 272 B = 17*16 B -> 16B-aligned rows, and the per-lane v2f
// fragment loads (offset row*68 + k + 2h) touch all 64 banks exactly once
// across a wave32 -> conflict-free.
#define LSTR 68

// ---------------------------------------------------------------------------
// 16B global->LDS copy: async (ASYNCcnt-tracked, no VGPR round trip) when the
// gfx1250 builtin exists, plain float4 staging otherwise.
#if defined(__AMDGCN__) && __has_builtin(__builtin_amdgcn_global_load_async_to_lds_b128)
#define USE_ASYNC_LDS 1
#else
#define USE_ASYNC_LDS 0
#endif

__device__ __forceinline__ void lds_copy16(const float* __restrict__ g, float* l) {
#if USE_ASYNC_LDS
  // Global source: AS(1), non-const, pointee = int vector_size(16).
  auto gp = (__attribute__((address_space(1))) v4i_raw*)g;
  // Generic pointers to LDS carry the LDS offset in their low 32 bits;
  // AS(3) pointers are 32-bit on AMDGPU.
  auto lp = (__attribute__((address_space(3))) v4i_raw*)(unsigned)(unsigned long long)l;
  __builtin_amdgcn_global_load_async_to_lds_b128(gp, lp, 0, 0);
#else
  *reinterpret_cast<float4*>(l) = *reinterpret_cast<const float4*>(g);
#endif
}

__device__ __forceinline__ void lds_copy_wait() {
#if USE_ASYNC_LDS
#if __has_builtin(__builtin_amdgcn_s_wait_asynccnt)
  __builtin_amdgcn_s_wait_asynccnt(0);
#else
  asm volatile("s_wait_asynccnt 0" ::: "memory");
#endif
#endif
}

// ---------------------------------------------------------------------------
// xm = x - b_dec
__global__ void k_sub_bias(const float* __restrict__ x,
                           const float* __restrict__ b_dec,
                           float* __restrict__ xm) {
  int i = blockIdx.x * blockDim.x + threadIdx.x;
  if (i < NB * ND) {
    int d = i - (i / ND) * ND;
    xm[i] = x[i] - b_dec[d];
  }
}

// ---------------------------------------------------------------------------
// WT[f*ND + d] = W[d*NF + f]   (W is [ND, NF])
__global__ void k_transpose(const float* __restrict__ W, float* __restrict__ WT) {
  __shared__ float tile[32][33];
  int f0 = blockIdx.x * 32, d0 = blockIdx.y * 32;
  int tx = threadIdx.x, ty = threadIdx.y;  // 32 x 8
#pragma unroll
  for (int i = 0; i < 4; ++i)
    tile[ty + i * 8][tx] = W[(size_t)(d0 + ty + i * 8) * NF + f0 + tx];
  __syncthreads();
#pragma unroll
  for (int i = 0; i < 4; ++i)
    WT[(size_t)(f0 + ty + i * 8) * ND + d0 + tx] = tile[tx][ty + i * 8];
}

// ---------------------------------------------------------------------------
// pre[b, f] = relu( sum_d xm[b,d] * W_enc[f,d] + b_enc[f] )
// Block: 512 threads = 16 waves, 128x128 output tile, waves in a 4x4 grid.
// Each wave register-blocks 2x2 WMMA tiles (32x32): 4 v_wmma per k-step off
// 2 A-frags + 2 B-frags -> 1 LDS fragment load per WMMA (vs 2 unblocked),
// matching LDS bandwidth to the fp32 WMMA issue rate.
__global__ __launch_bounds__(512) void k_encode(const float* __restrict__ xm,
                                                const float* __restrict__ W_enc,
                                                const float* __restrict__ b_enc,
                                                float* __restrict__ pre) {
  __shared__ float as[128 * LSTR];
  __shared__ float bs[128 * LSTR];
  const int f0g = blockIdx.x * 128;
  const int b0g = blockIdx.y * 128;
  const int tid = threadIdx.x;
  const int wave = tid >> 5, lane = tid & 31;
  const int wm = wave >> 2, wn = wave & 3;   // 4x4 grid of 32x32 wave tiles
  const int h = lane >> 4, r = lane & 15;

  v8f acc[2][2] = {};
  for (int kc = 0; kc < ND; kc += 64) {
#pragma unroll
    for (int it = 0; it < 4; ++it) {
      int idx = tid + it * 512;
      int row = idx >> 4, q = idx & 15;     // row 0..127, q 0..15 (float4 col)
      lds_copy16(&xm[(size_t)(b0g + row) * ND + kc + q * 4],
                 &as[row * LSTR + q * 4]);
      lds_copy16(&W_enc[(size_t)(f0g + row) * ND + kc + q * 4],
                 &bs[row * LSTR + q * 4]);
    }
    lds_copy_wait();
    __syncthreads();
#pragma unroll
    for (int kk = 0; kk < 64; kk += 4) {
      // A 16x4: lanes 0-15 -> K=kk,kk+1 ; lanes 16-31 -> K=kk+2,kk+3
      v2f a0 = *reinterpret_cast<const v2f*>(
          &as[(wm * 32 + r) * LSTR + kk + 2 * h]);
      v2f a1 = *reinterpret_cast<const v2f*>(
          &as[(wm * 32 + 16 + r) * LSTR + kk + 2 * h]);
      v2f b0 = *reinterpret_cast<const v2f*>(
          &bs[(wn * 32 + r) * LSTR + kk + 2 * h]);
      v2f b1 = *reinterpret_cast<const v2f*>(
          &bs[(wn * 32 + 16 + r) * LSTR + kk + 2 * h]);
      acc[0][0] = __builtin_amdgcn_wmma_f32_16x16x4_f32(false, a0, false, b0,
                                                        (short)0, acc[0][0], false, false);
      acc[0][1] = __builtin_amdgcn_wmma_f32_16x16x4_f32(false, a0, false, b1,
                                                        (short)0, acc[0][1], false, false);
      acc[1][0] = __builtin_amdgcn_wmma_f32_16x16x4_f32(false, a1, false, b0,
                                                        (short)0, acc[1][0], false, false);
      acc[1][1] = __builtin_amdgcn_wmma_f32_16x16x4_f32(false, a1, false, b1,
                                                        (short)0, acc[1][1], false, false);
    }
    __syncthreads();
  }
  // C/D layout: VGPR r8 -> M = r8 (lanes 0-15) or r8+8 (lanes 16-31), N = lane&15
#pragma unroll
  for (int j = 0; j < 2; ++j) {
    const int n = wn * 32 + j * 16 + r;
    const float be = b_enc[f0g + n];
#pragma unroll
    for (int i = 0; i < 2; ++i) {
#pragma unroll
      for (int r8 = 0; r8 < 8; ++r8) {
        int m = wm * 32 + i * 16 + r8 + 8 * h;
        float v = acc[i][j][r8] + be;
        v = v > 0.f ? v : 0.f;
        pre[(size_t)(b0g + m) * NF + f0g + n] = v;
      }
    }
  }
}

// ---------------------------------------------------------------------------
// Per-row top-64 via 64 rounds of argmax over an LDS-resident row (96 KB).
// Ties resolved to lowest index (matches jax.lax.top_k ordering semantics).
__global__ __launch_bounds__(256) void k_topk(const float* __restrict__ pre,
                                              float* __restrict__ vals,
                                              int* __restrict__ idx) {
  __shared__ float srow[NF];
  __shared__ float smax[256];
  __shared__ int simax[256];
  const int b = blockIdx.x, t = threadIdx.x;
  for (int i = t; i < NF; i += 256) srow[i] = pre[(size_t)b * NF + i];
  __syncthreads();
  for (int k = 0; k < NK; ++k) {
    float mv = -1e30f;
    int mi = NF;
    for (int i = t; i < NF; i += 256) {
      float v = srow[i];
      if (v > mv) { mv = v; mi = i; }
    }
    smax[t] = mv;
    simax[t] = mi;
    __syncthreads();
    for (int s = 128; s > 0; s >>= 1) {
      if (t < s) {
        float ov = smax[t + s];
        int oi = simax[t + s];
        if (ov > smax[t] || (ov == smax[t] && oi < simax[t])) {
          smax[t] = ov;
          simax[t] = oi;
        }
      }
      __syncthreads();
    }
    if (t == 0) {
      vals[(size_t)b * NK + k] = smax[0];
      idx[(size_t)b * NK + k] = simax[0];
      srow[simax[0]] = -1e30f;
    }
    __syncthreads();
  }
}

// ---------------------------------------------------------------------------
// Per token b: vw[kd,ku] = <W_enc_down[down_idx[kd]], W_dec_up.T[up_idx[ku]]>
// approx[kd] = sum_ku vw * count(connections) * up_vals[ku]
// 64x64x768 WMMA GEMM + multiplicity epilogue; deterministic LDS reduction.
__global__ __launch_bounds__(512) void k_contrib(
    const float* __restrict__ W_enc_down, const float* __restrict__ WduT,
    const long long* __restrict__ connections, const int* __restrict__ up_idx,
    const float* __restrict__ up_vals, const int* __restrict__ down_idx,
    float* __restrict__ approx) {
  __shared__ float as[64 * LSTR];
  __shared__ float bs[64 * LSTR];
  __shared__ int s_didx[NK], s_uidx[NK];
  __shared__ float s_uvals[NK];
  __shared__ int s_conn[NK * NC];
  __shared__ float s_part[NK * 4];

  const int b = blockIdx.x, tid = threadIdx.x;
  const int wave = tid >> 5, lane = tid & 31;
  const int tm = wave >> 2, tn = wave & 3;
  const int h = lane >> 4, r = lane & 15;

  if (tid < NK) {
    s_didx[tid] = down_idx[(size_t)b * NK + tid];
    s_uidx[tid] = up_idx[(size_t)b * NK + tid];
    s_uvals[tid] = up_vals[(size_t)b * NK + tid];
  }
  __syncthreads();
  for (int i = tid; i < NK * NC; i += 512) {
    int rowi = i >> 5, c = i & 31;
    s_conn[i] = (int)connections[(size_t)s_didx[rowi] * NC + c];
  }
  // (s_conn is only read after the syncthreads inside the GEMM loop)

  v8f acc = {};
  for (int kc = 0; kc < ND; kc += 64) {
#pragma unroll
    for (int it = 0; it < 2; ++it) {
      int idx = tid + it * 512;
      int row = idx >> 4, q = idx & 15;
      lds_copy16(&W_enc_down[(size_t)s_didx[row] * ND + kc + q * 4],
                 &as[row * LSTR + q * 4]);
      lds_copy16(&WduT[(size_t)s_uidx[row] * ND + kc + q * 4],
                 &bs[row * LSTR + q * 4]);
    }
    lds_copy_wait();
    __syncthreads();
#pragma unroll
    for (int kk = 0; kk < 64; kk += 4) {
      v2f af = *reinterpret_cast<const v2f*>(&as[(tm * 16 + r) * LSTR + kk + 2 * h]);
      v2f bf = *reinterpret_cast<const v2f*>(&bs[(tn * 16 + r) * LSTR + kk + 2 * h]);
      acc = __builtin_amdgcn_wmma_f32_16x16x4_f32(false, af, false, bf,
                                                  (short)0, acc, false, false);
    }
    __syncthreads();
  }

  // Epilogue: multiply by connection multiplicity and upstream value,
  // reduce over ku (lanes 0-15 / 16-31 share the same kd per VGPR).
  const int ku = tn * 16 + r;
  const int ufeat = s_uidx[ku];
  const float uval = s_uvals[ku];
#pragma unroll
  for (int r8 = 0; r8 < 8; ++r8) {
    int kd = tm * 16 + r8 + 8 * h;
    int cnt = 0;
    const int* crow = &s_conn[kd * NC];
#pragma unroll 8
    for (int c = 0; c < NC; ++c) cnt += (crow[c] == ufeat) ? 1 : 0;
    float p = acc[r8] * (float)cnt * uval;
    p += __shfl_xor(p, 1);
    p += __shfl_xor(p, 2);
    p += __shfl_xor(p, 4);
    p += __shfl_xor(p, 8);
    if (r == 0) s_part[kd * 4 + tn] = p;  // unique (kd, tn) slot: deterministic
  }
  __syncthreads();
  if (tid < NK) {
    float s = s_part[tid * 4 + 0] + s_part[tid * 4 + 1] +
              s_part[tid * 4 + 2] + s_part[tid * 4 + 3];
    approx[(size_t)b * NK + tid] = s;
  }
}

// ---------------------------------------------------------------------------
// out[b, :] = b_dec + sum_k vals[b,k] * WT[idx[b,k], :]   (WT is [NF, ND])
__global__ __launch_bounds__(256) void k_recon(const float* __restrict__ WT,
                                               const float* __restrict__ b_dec,
                                               const float* __restrict__ vals,
                                               const int* __restrict__ idx,
                                               float* __restrict__ out) {
  __shared__ float sv[NK];
  __shared__ int si[NK];
  const int b = blockIdx.x, t = threadIdx.x;
  if (t < NK) {
    sv[t] = vals[(size_t)b * NK + t];
    si[t] = idx[(size_t)b * NK + t];
  }
  __syncthreads();
  float a0 = 0.f, a1 = 0.f, a2 = 0.f;
  const int d0 = t, d1 = t + 256, d2 = t + 512;
  for (int k = 0; k < NK; ++k) {
    const float* row = WT + (size_t)si[k] * ND;
    float v = sv[k];
    a0 += v * row[d0];
    a1 += v * row[d1];
    a2 += v * row[d2];
  }
  out[(size_t)b * ND + d0] = a0 + b_dec[d0];
  out[(size_t)b * ND + d1] = a1 + b_dec[d1];
  out[(size_t)b * ND + d2] = a2 + b_dec[d2];
}

// ---------------------------------------------------------------------------
extern "C" void kernel_launch(void* const* d_in, const int* in_sizes, int n_in,
                              void* d_out, int out_size, void* d_ws,
                              size_t ws_size, hipStream_t stream) {
  (void)in_sizes; (void)n_in; (void)out_size; (void)ws_size;
  const float* x_up       = (const float*)d_in[0];
  const float* x_down     = (const float*)d_in[1];
  const float* W_enc_up   = (const float*)d_in[2];
  const float* b_enc_up   = (const float*)d_in[3];
  const float* W_dec_up   = (const float*)d_in[4];
  const float* b_dec_up   = (const float*)d_in[5];
  const float* W_enc_down = (const float*)d_in[6];
  const float* b_enc_down = (const float*)d_in[7];
  const float* W_dec_down = (const float*)d_in[8];
  const float* b_dec_down = (const float*)d_in[9];
  const long long* connections = (const long long*)d_in[10];
  float* out = (float*)d_out;

  char* p = (char*)d_ws;
  auto take = [&](size_t bytes) -> void* {
    void* r = (void*)p;
    p += (bytes + 255) & ~(size_t)255;
    return r;
  };
  float* WduT   = (float*)take((size_t)NF * ND * 4);  // W_dec_up transposed
  float* WddT   = (float*)take((size_t)NF * ND * 4);  // W_dec_down transposed
  float* pre    = (float*)take((size_t)NB * NF * 4);  // reused up then down
  float* xm_up  = (float*)take((size_t)NB * ND * 4);
  float* xm_dn  = (float*)take((size_t)NB * ND * 4);
  float* up_v   = (float*)take((size_t)NB * NK * 4);
  int*   up_i   = (int*)take((size_t)NB * NK * 4);
  float* dn_v   = (float*)take((size_t)NB * NK * 4);
  int*   dn_i   = (int*)take((size_t)NB * NK * 4);
  float* approx = (float*)take((size_t)NB * NK * 4);

  const dim3 tgrid(NF / 32, ND / 32), tblk(32, 8);
  const dim3 egrid(NF / 128, NB / 128);

  k_sub_bias<<<(NB * ND + 255) / 256, 256, 0, stream>>>(x_up, b_dec_up, xm_up);
  k_sub_bias<<<(NB * ND + 255) / 256, 256, 0, stream>>>(x_down, b_dec_down, xm_dn);
  k_transpose<<<tgrid, tblk, 0, stream>>>(W_dec_up, WduT);
  k_transpose<<<tgrid, tblk, 0, stream>>>(W_dec_down, WddT);

  k_encode<<<egrid, 512, 0, stream>>>(xm_up, W_enc_up, b_enc_up, pre);
  k_topk<<<NB, 256, 0, stream>>>(pre, up_v, up_i);
  k_encode<<<egrid, 512, 0, stream>>>(xm_dn, W_enc_down, b_enc_down, pre);
  k_topk<<<NB, 256, 0, stream>>>(pre, dn_v, dn_i);

  k_contrib<<<NB, 512, 0, stream>>>(W_enc_down, WduT, connections, up_i, up_v,
                                    dn_i, approx);

  k_recon<<<NB, 256, 0, stream>>>(WduT, b_dec_up, up_v, up_i, out);
  k_recon<<<NB, 256, 0, stream>>>(WddT, b_dec_down, approx, dn_i,
                                  out + (size_t)NB * ND);
}